// TensorCPClassifier_82154134438095
// MI455X (gfx1250) — compile-verified
//
#include <hip/hip_runtime.h>
#include <hip/hip_bf16.h>
#include <math.h>
#include <stdint.h>

// ---------------------------------------------------------------------------
// MI455X / gfx1250 implementation. wave32. WMMA f32 16x16x4 for all GEMMs.
// Async global->LDS staging (ASYNCcnt) for the Gram K-slices.
// ---------------------------------------------------------------------------

typedef __attribute__((ext_vector_type(2))) float v2f;
typedef __attribute__((ext_vector_type(8))) float v8f;

__device__ __forceinline__ v8f wmma4(v2f a, v2f b, v8f c) {
  // D = A(16x4) * B(4x16) + C(16x16), all f32, wave32.
  return __builtin_amdgcn_wmma_f32_16x16x4_f32(
      /*neg_a=*/false, a, /*neg_b=*/false, b,
      /*c_mod=*/(short)0, c, /*reuse_a=*/false, /*reuse_b=*/false);
}

// Low 32 bits of a generic pointer to __shared__ = LDS byte address.
__device__ __forceinline__ uint32_t lds_addr32(const void* p) {
  return (uint32_t)(uintptr_t)p;
}

// ---------------------------------------------------------------------------
// Kernel A: per-batch CP-ALS.  One 256-thread block (8 waves) per batch.
// Tensor streamed from global (151MB total -> resident in 192MB L2).
// ---------------------------------------------------------------------------

#define CPR 4

// Gram of mode-MODE unfolding: sG = Un * Un^T (64x64, K = 64 slices x 9).
// Each slice (9 K-values, zero-padded to 12) is staged straight into LDS with
// CDNA5 GLOBAL_LOAD_ASYNC_TO_LDS_B32 (no VGPR round-trip; ASYNCcnt tracked),
// then all 8 waves run 16x16x4 WMMAs out of LDS with per-wave register
// accumulators (2 tiles per wave).
template <int MODE>
__device__ void gram_accum(const float* __restrict__ Tb,
                           float (*sSlice)[12], float (*sG)[64],
                           int tid, int wav, int half, int r15) {
  const int t0 = wav, t1 = wav + 8;
  const int rA0 = (t0 >> 2) * 16 + r15, cB0 = (t0 & 3) * 16 + r15;
  const int rA1 = (t1 >> 2) * 16 + r15, cB1 = (t1 & 3) * 16 + r15;
  // zero the K-pad columns once (never written by staging)
  if (tid < 64) { sSlice[tid][9] = 0.f; sSlice[tid][10] = 0.f; sSlice[tid][11] = 0.f; }
  // 2-D staging map: thread (q = row 0..63, j = 0..3) stages elements
  // j, j+4 (and j==0 also stages element 8) of row q.  No divisions.
  const int q = tid >> 2, j = tid & 3;
  const uint32_t lds0 = lds_addr32(&sSlice[q][j]);
  const uint32_t lds4 = lds_addr32(&sSlice[q][j + 4]);
  const uint32_t lds8 = lds_addr32(&sSlice[q][8]);
  v8f c0 = {}, c1 = {};
  for (int sl = 0; sl < 64; ++sl) {
    const uint32_t gbase = (MODE == 0)
        ? (uint32_t)(q * 576 + sl * 9) * 4u    // Un0[a][(sl,cd)]
        : (uint32_t)(sl * 576 + q * 9) * 4u;   // Un1[b][(sl,cd)]
    __syncthreads();  // previous slice fully consumed
    asm volatile("global_load_async_to_lds_b32 %0, %1, %2"
                 :: "v"(lds0), "v"(gbase + 4u * (uint32_t)j), "s"(Tb)
                 : "memory");
    asm volatile("global_load_async_to_lds_b32 %0, %1, %2"
                 :: "v"(lds4), "v"(gbase + 4u * (uint32_t)(j + 4)), "s"(Tb)
                 : "memory");
    if (j == 0)
      asm volatile("global_load_async_to_lds_b32 %0, %1, %2"
                   :: "v"(lds8), "v"(gbase + 32u), "s"(Tb)
                   : "memory");
    __builtin_prefetch(Tb + (sl + 1) * 576, 0, 1);
    asm volatile("s_wait_asynccnt 0x0" ::: "memory");
    __syncthreads();
#pragma unroll
    for (int kc = 0; kc < 12; kc += 4) {
      int ka = kc + 2 * half;
      v2f a0 = *(const v2f*)&sSlice[rA0][ka];
      v2f b0 = *(const v2f*)&sSlice[cB0][ka];
      c0 = wmma4(a0, b0, c0);
      v2f a1 = *(const v2f*)&sSlice[rA1][ka];
      v2f b1 = *(const v2f*)&sSlice[cB1][ka];
      c1 = wmma4(a1, b1, c1);
    }
  }
  __syncthreads();
#pragma unroll
  for (int i = 0; i < 8; ++i) {
    sG[(t0 >> 2) * 16 + i + 8 * half][cB0] = c0[i];
    sG[(t1 >> 2) * 16 + i + 8 * half][cB1] = c1[i];
  }
}

// 4x4 gram of rows x 4 factor
__device__ void gram4(const float* F, int rows, float g[4][4]) {
  for (int s = 0; s < 4; ++s)
    for (int r = 0; r < 4; ++r) g[s][r] = 0.f;
  for (int i = 0; i < rows; ++i)
    for (int s = 0; s < 4; ++s)
      for (int r = 0; r < 4; ++r) g[s][r] += F[i * 4 + s] * F[i * 4 + r];
}

// Vinv = inverse of hadamard(gram(FA),gram(FB),gram(FC)).  Thread-0 only.
__device__ void make_vinv(const float* FA, int ra, const float* FB, int rb,
                          const float* FC, int rc, float* Vinv /*16*/) {
  float gA[4][4], gB[4][4], gC[4][4], V[4][4], inv[4][4];
  gram4(FA, ra, gA);
  gram4(FB, rb, gB);
  gram4(FC, rc, gC);
  for (int s = 0; s < 4; ++s)
    for (int r = 0; r < 4; ++r) {
      V[s][r] = gA[s][r] * gB[s][r] * gC[s][r];
      inv[s][r] = (s == r) ? 1.f : 0.f;
    }
  for (int i = 0; i < 4; ++i) V[i][i] += 1e-6f;
  for (int col = 0; col < 4; ++col) {
    int piv = col;
    for (int r2 = col + 1; r2 < 4; ++r2)
      if (fabsf(V[r2][col]) > fabsf(V[piv][col])) piv = r2;
    if (piv != col)
      for (int j = 0; j < 4; ++j) {
        float t = V[col][j]; V[col][j] = V[piv][j]; V[piv][j] = t;
        t = inv[col][j]; inv[col][j] = inv[piv][j]; inv[piv][j] = t;
      }
    float p = V[col][col];
    if (fabsf(p) < 1e-20f) p = 1e-20f;
    float ip = 1.f / p;
    for (int j = 0; j < 4; ++j) { V[col][j] *= ip; inv[col][j] *= ip; }
    for (int r2 = 0; r2 < 4; ++r2) {
      if (r2 == col) continue;
      float f = V[r2][col];
      for (int j = 0; j < 4; ++j) {
        V[r2][j] -= f * V[col][j];
        inv[r2][j] -= f * inv[col][j];
      }
    }
  }
  for (int s = 0; s < 4; ++s)
    for (int r = 0; r < 4; ++r) Vinv[s * 4 + r] = inv[s][r];
}

// top-3 eigenvectors of 3x3 SPD G via subspace iteration -> factor 3x4
// (4th column = deterministic pad, mirroring the random pad in the reference)
__device__ void eig3_to_factor(const float* G, float* F /*3x4*/, int n) {
  float Q[3][3] = {{1.f, 0.f, 0.f}, {0.f, 1.f, 0.f}, {0.f, 0.f, 1.f}};
  for (int it = 0; it < 12; ++it) {
    float Y[3][3];
    for (int i = 0; i < 3; ++i)
      for (int c = 0; c < 3; ++c) {
        float s = 0.f;
        for (int j = 0; j < 3; ++j) s += G[i * 3 + j] * Q[j][c];
        Y[i][c] = s;
      }
    for (int c = 0; c < 3; ++c) {
      float nr = 1e-20f;
      for (int i = 0; i < 3; ++i) nr += Y[i][c] * Y[i][c];
      nr = rsqrtf(nr);
      for (int i = 0; i < 3; ++i) Y[i][c] *= nr;
      for (int s2 = c + 1; s2 < 3; ++s2) {
        float d = 0.f;
        for (int i = 0; i < 3; ++i) d += Y[i][s2] * Y[i][c];
        for (int i = 0; i < 3; ++i) Y[i][s2] -= d * Y[i][c];
      }
    }
    for (int i = 0; i < 3; ++i)
      for (int c = 0; c < 3; ++c) Q[i][c] = Y[i][c];
  }
  for (int i = 0; i < 3; ++i) {
    for (int c = 0; c < 3; ++c) F[i * 4 + c] = Q[i][c];
    F[i * 4 + 3] = __sinf(1.234f * (float)(i + 1) * (float)(n + 1));
  }
}

__global__ __launch_bounds__(256) void cp_als_kernel(
    const float* __restrict__ Tin, float* __restrict__ combined) {
  __shared__ float sG[64][64];           // 16 KB: unfolding gram
  __shared__ float sSlice[64][12];       // 3 KB: staged K-slice
  __shared__ float sQ[64][CPR];
  __shared__ float sY[64][CPR];
  __shared__ float sM[64][CPR];          // MTTKRP result
  __shared__ float sF0[64][CPR], sF1[64][CPR];
  __shared__ float sF2[3][CPR], sF3[3][CPR];
  __shared__ float sW[CPR][9];           // W[r][c*3+d] = F2[c][r]*F3[d][r]
  __shared__ float sT2[3][3][CPR];       // tmp2[c][d][r]
  __shared__ float sVinv[16];
  __shared__ float sG2[3][3], sG3[3][3];

  const int tid  = threadIdx.x;
  const int lane = tid & 31;
  const int wav  = tid >> 5;
  const int half = lane >> 4;
  const int r15  = lane & 15;
  const float* Tb = Tin + (size_t)blockIdx.x * 36864;

  // ===== init: approx top-4 left singular vectors of mode-0/1 unfoldings ====
  for (int mode = 0; mode < 2; ++mode) {
    __syncthreads();
    if (mode == 0) gram_accum<0>(Tb, sSlice, sG, tid, wav, half, r15);
    else           gram_accum<1>(Tb, sSlice, sG, tid, wav, half, r15);
    __syncthreads();
    // subspace iteration for top-4 eigenvectors of sG
    if (tid < 64 * CPR) {
      sQ[tid >> 2][tid & 3] =
          __sinf(0.731f * (float)tid + 0.37f * (float)mode + 0.2f);
    }
    __syncthreads();
    for (int it = 0; it < 8; ++it) {
      if (tid < 64) {
        float y[CPR] = {0.f, 0.f, 0.f, 0.f};
        for (int j = 0; j < 64; ++j) {
          float g = sG[tid][j];
          for (int r = 0; r < CPR; ++r) y[r] += g * sQ[j][r];
        }
        for (int r = 0; r < CPR; ++r) sY[tid][r] = y[r];
      }
      __syncthreads();
      if (tid == 0) {  // modified Gram-Schmidt on columns of sY
        for (int r = 0; r < CPR; ++r) {
          float nr = 1e-20f;
          for (int i = 0; i < 64; ++i) nr += sY[i][r] * sY[i][r];
          nr = rsqrtf(nr);
          for (int i = 0; i < 64; ++i) sY[i][r] *= nr;
          for (int s = r + 1; s < CPR; ++s) {
            float d = 0.f;
            for (int i = 0; i < 64; ++i) d += sY[i][s] * sY[i][r];
            for (int i = 0; i < 64; ++i) sY[i][s] -= d * sY[i][r];
          }
        }
      }
      __syncthreads();
      if (tid < 64 * CPR) sQ[tid >> 2][tid & 3] = sY[tid >> 2][tid & 3];
      __syncthreads();
    }
    if (tid < 64 * CPR) {
      float v = sQ[tid >> 2][tid & 3];
      if (mode == 0) sF0[tid >> 2][tid & 3] = v;
      else           sF1[tid >> 2][tid & 3] = v;
    }
  }
  __syncthreads();

  // ===== 3x3 grams for modes 2,3; eigvec init + pad column =====
  if (tid < 9) { sG2[tid / 3][tid % 3] = 0.f; sG3[tid / 3][tid % 3] = 0.f; }
  __syncthreads();
  {
    float a2[9], a3[9];
    for (int i = 0; i < 9; ++i) { a2[i] = 0.f; a3[i] = 0.f; }
    for (int p = tid; p < 4096; p += 256) {
      const float* tp = Tb + p * 9;
      __builtin_prefetch(tp + 9 * 256, 0, 1);
      float v[9];
      for (int i = 0; i < 9; ++i) v[i] = tp[i];
      for (int c = 0; c < 3; ++c)
        for (int c2 = 0; c2 < 3; ++c2) {
          float s = 0.f;
          for (int d = 0; d < 3; ++d) s += v[c * 3 + d] * v[c2 * 3 + d];
          a2[c * 3 + c2] += s;
        }
      for (int d = 0; d < 3; ++d)
        for (int d2 = 0; d2 < 3; ++d2) {
          float s = 0.f;
          for (int c = 0; c < 3; ++c) s += v[c * 3 + d] * v[c * 3 + d2];
          a3[d * 3 + d2] += s;
        }
    }
    for (int i = 0; i < 9; ++i) {
      atomicAdd(&sG2[i / 3][i % 3], a2[i]);
      atomicAdd(&sG3[i / 3][i % 3], a3[i]);
    }
  }
  __syncthreads();
  if (tid == 0) eig3_to_factor(&sG2[0][0], &sF2[0][0], 2);
  if (tid == 1) eig3_to_factor(&sG3[0][0], &sF3[0][0], 3);
  __syncthreads();

  // ===== ALS sweeps =====
  for (int iter = 0; iter < 10; ++iter) {
    // W[r][cd] = F2[c][r]*F3[d][r]  (F2,F3 fixed through modes 0,1)
    if (tid < 36) {
      int r = tid / 9, cd = tid % 9;
      sW[r][cd] = sF2[cd / 3][r] * sF3[cd % 3][r];
    }
    if (tid < 256) (&sM[0][0])[tid] = 0.f;
    __syncthreads();

    // ---- mode 0: M0(a,r) = sum_b F1(b,r) * sum_cd T[a,b,cd] W[r][cd] ----
    {
      int a = tid >> 2, q = tid & 3;
      float acc[4] = {0.f, 0.f, 0.f, 0.f};
      for (int b = q * 16; b < q * 16 + 16; ++b) {
        const float* tp = Tb + (a * 64 + b) * 9;
        float v[9];
        for (int i = 0; i < 9; ++i) v[i] = tp[i];
        for (int r = 0; r < 4; ++r) {
          float tv = 0.f;
          for (int cd = 0; cd < 9; ++cd) tv += v[cd] * sW[r][cd];
          acc[r] += tv * sF1[b][r];
        }
      }
      for (int r = 0; r < 4; ++r) atomicAdd(&sM[a][r], acc[r]);
    }
    __syncthreads();
    if (tid == 0) make_vinv(&sF1[0][0], 64, &sF2[0][0], 3, &sF3[0][0], 3, sVinv);
    __syncthreads();
    if (tid < 64) {
      float o[4];
      for (int r = 0; r < 4; ++r) {
        float s = 0.f;
        for (int x = 0; x < 4; ++x) s += sM[tid][x] * sVinv[x * 4 + r];
        o[r] = s;
      }
      for (int r = 0; r < 4; ++r) sF0[tid][r] = o[r];
    }
    if (tid < 256) (&sM[0][0])[tid] = 0.f;
    __syncthreads();

    // ---- mode 1 ----
    {
      int b = tid >> 2, q = tid & 3;
      float acc[4] = {0.f, 0.f, 0.f, 0.f};
      for (int a = q * 16; a < q * 16 + 16; ++a) {
        const float* tp = Tb + (a * 64 + b) * 9;
        float v[9];
        for (int i = 0; i < 9; ++i) v[i] = tp[i];
        for (int r = 0; r < 4; ++r) {
          float tv = 0.f;
          for (int cd = 0; cd < 9; ++cd) tv += v[cd] * sW[r][cd];
          acc[r] += tv * sF0[a][r];
        }
      }
      for (int r = 0; r < 4; ++r) atomicAdd(&sM[b][r], acc[r]);
    }
    __syncthreads();
    if (tid == 0) make_vinv(&sF0[0][0], 64, &sF2[0][0], 3, &sF3[0][0], 3, sVinv);
    __syncthreads();
    if (tid < 64) {
      float o[4];
      for (int r = 0; r < 4; ++r) {
        float s = 0.f;
        for (int x = 0; x < 4; ++x) s += sM[tid][x] * sVinv[x * 4 + r];
        o[r] = s;
      }
      for (int r = 0; r < 4; ++r) sF1[tid][r] = o[r];
    }
    if (tid < 36) (&sT2[0][0][0])[tid] = 0.f;
    __syncthreads();

    // ---- tmp2[c][d][r] = sum_ab T[a,b,c,d] F0(a,r) F1(b,r)  (modes 2&3) ----
    {
      float acc[36];
      for (int i = 0; i < 36; ++i) acc[i] = 0.f;
      for (int p = tid; p < 4096; p += 256) {
        int a = p >> 6, b = p & 63;
        const float* tp = Tb + p * 9;
        float v[9];
        for (int i = 0; i < 9; ++i) v[i] = tp[i];
        float ab[4];
        for (int r = 0; r < 4; ++r) ab[r] = sF0[a][r] * sF1[b][r];
        for (int cd = 0; cd < 9; ++cd)
          for (int r = 0; r < 4; ++r) acc[cd * 4 + r] += v[cd] * ab[r];
      }
      for (int i = 0; i < 36; ++i) atomicAdd(&(&sT2[0][0][0])[i], acc[i]);
    }
    __syncthreads();

    // ---- mode 2 ----
    if (tid < 12) {
      int c = tid >> 2, r = tid & 3;
      float s = 0.f;
      for (int d = 0; d < 3; ++d) s += sT2[c][d][r] * sF3[d][r];
      sM[c][r] = s;
    }
    __syncthreads();
    if (tid == 0) make_vinv(&sF0[0][0], 64, &sF1[0][0], 64, &sF3[0][0], 3, sVinv);
    __syncthreads();
    if (tid < 3) {
      float o[4];
      for (int r = 0; r < 4; ++r) {
        float s = 0.f;
        for (int x = 0; x < 4; ++x) s += sM[tid][x] * sVinv[x * 4 + r];
        o[r] = s;
      }
      for (int r = 0; r < 4; ++r) sF2[tid][r] = o[r];
    }
    __syncthreads();

    // ---- mode 3 ----
    if (tid < 12) {
      int d = tid >> 2, r = tid & 3;
      float s = 0.f;
      for (int c = 0; c < 3; ++c) s += sT2[c][d][r] * sF2[c][r];
      sM[d][r] = s;
    }
    __syncthreads();
    if (tid == 0) make_vinv(&sF0[0][0], 64, &sF1[0][0], 64, &sF2[0][0], 3, sVinv);
    __syncthreads();
    if (tid < 3) {
      float o[4];
      for (int r = 0; r < 4; ++r) {
        float s = 0.f;
        for (int x = 0; x < 4; ++x) s += sM[tid][x] * sVinv[x * 4 + r];
        o[r] = s;
      }
      for (int r = 0; r < 4; ++r) sF3[tid][r] = o[r];
    }
    __syncthreads();
  }

  // ===== emit cp features: [f0(256) | f1(256) | f2(12) | f3(12)] =====
  float* out = combined + (size_t)blockIdx.x * 664;
  if (tid < 256) {
    out[tid]       = (&sF0[0][0])[tid];
    out[256 + tid] = (&sF1[0][0])[tid];
  }
  if (tid < 12) {
    out[512 + tid] = (&sF2[0][0])[tid];
    out[524 + tid] = (&sF3[0][0])[tid];
  }
}

// ---------------------------------------------------------------------------
// MLP kernels: 16-row batch tile per block, 4 waves = 4 N-tiles of 16.
// ---------------------------------------------------------------------------

__global__ __launch_bounds__(128) void seed_mlp_kernel(
    const float* __restrict__ ic, const float* __restrict__ sir,
    const float* __restrict__ icc, const float* __restrict__ sirc,
    const float* __restrict__ w1, const float* __restrict__ b1,
    const float* __restrict__ w2, const float* __restrict__ b2,
    float* __restrict__ combined) {
  __shared__ float sAct[16][64];
  const int tid = threadIdx.x;
  const int lane = tid & 31, w = tid >> 5;
  const int half = lane >> 4, r = lane & 15;
  const int m0 = blockIdx.x * 16;
  const int colB = w * 16 + r;
  const int rowA = m0 + r;
  const float* bufs[4] = {ic, sir, icc, sirc};

  v8f acc = {};
#pragma unroll
  for (int seg = 0; seg < 4; ++seg) {  // 4 input buffers x 64 features
    const float* xp = bufs[seg] + (size_t)rowA * 64;
    for (int k0 = 0; k0 < 64; k0 += 4) {
      int ka = k0 + 2 * half;
      v2f a = *(const v2f*)(xp + ka);
      v2f b;
      int kw = seg * 64 + ka;
      b[0] = w1[kw * 64 + colB];
      b[1] = w1[(kw + 1) * 64 + colB];
      acc = wmma4(a, b, acc);
    }
  }
  for (int i = 0; i < 8; ++i) {
    float v = acc[i] + b1[colB];
    sAct[i + 8 * half][colB] = v > 0.f ? v : 0.f;
  }
  __syncthreads();

  acc = (v8f){};
  for (int k0 = 0; k0 < 64; k0 += 4) {
    int ka = k0 + 2 * half;
    v2f a = *(const v2f*)&sAct[r][ka];
    v2f b;
    b[0] = w2[ka * 64 + colB];
    b[1] = w2[(ka + 1) * 64 + colB];
    acc = wmma4(a, b, acc);
  }
  for (int i = 0; i < 8; ++i)
    combined[(size_t)(m0 + i + 8 * half) * 664 + 536 + colB] = acc[i] + b2[colB];
}

__global__ __launch_bounds__(128) void scal_mlp_kernel(
    const float* __restrict__ xs, const float* __restrict__ w1,
    const float* __restrict__ b1, const float* __restrict__ w2,
    const float* __restrict__ b2, float* __restrict__ combined) {
  __shared__ float sAct[16][64];
  const int tid = threadIdx.x;
  const int lane = tid & 31, w = tid >> 5;
  const int half = lane >> 4, r = lane & 15;
  const int m0 = blockIdx.x * 16;
  const int colB = w * 16 + r;
  const int rowA = m0 + r;

  v8f acc = {};
  {  // K = 4: single WMMA
    int ka = 2 * half;
    v2f a = *(const v2f*)(xs + (size_t)rowA * 4 + ka);
    v2f b;
    b[0] = w1[ka * 64 + colB];
    b[1] = w1[(ka + 1) * 64 + colB];
    acc = wmma4(a, b, acc);
  }
  for (int i = 0; i < 8; ++i) {
    float v = acc[i] + b1[colB];
    sAct[i + 8 * half][colB] = v > 0.f ? v : 0.f;
  }
  __syncthreads();

  acc = (v8f){};
  for (int k0 = 0; k0 < 64; k0 += 4) {
    int ka = k0 + 2 * half;
    v2f a = *(const v2f*)&sAct[r][ka];
    v2f b;
    b[0] = w2[ka * 64 + colB];
    b[1] = w2[(ka + 1) * 64 + colB];
    acc = wmma4(a, b, acc);
  }
  for (int i = 0; i < 8; ++i)
    combined[(size_t)(m0 + i + 8 * half) * 664 + 600 + colB] = acc[i] + b2[colB];
}

__global__ __launch_bounds__(128) void cls_kernel(
    const float* __restrict__ combined, const float* __restrict__ w1,
    const float* __restrict__ b1, const float* __restrict__ w2,
    const float* __restrict__ b2, float* __restrict__ outp) {
  __shared__ float sAct[16][64];
  const int tid = threadIdx.x;
  const int lane = tid & 31, w = tid >> 5;
  const int half = lane >> 4, r = lane & 15;
  const int m0 = blockIdx.x * 16;
  const int colB = w * 16 + r;
  const int rowA = m0 + r;
  const float* xrow = combined + (size_t)rowA * 664;

  v8f acc = {};
  for (int k0 = 0; k0 < 664; k0 += 4) {  // 664 = 166 * 4
    int ka = k0 + 2 * half;
    v2f a = *(const v2f*)(xrow + ka);
    v2f b;
    b[0] = w1[ka * 64 + colB];
    b[1] = w1[(ka + 1) * 64 + colB];
    acc = wmma4(a, b, acc);
  }
  for (int i = 0; i < 8; ++i) {
    float v = acc[i] + b1[colB];
    sAct[i + 8 * half][colB] = v > 0.f ? v : 0.f;
  }
  __syncthreads();

  if (tid < 16) {
    float s = b2[0];
    for (int k = 0; k < 64; ++k) s += sAct[tid][k] * w2[k];
    outp[m0 + tid] = s;
  }
}

// ---------------------------------------------------------------------------

extern "C" void kernel_launch(void* const* d_in, const int* in_sizes, int n_in,
                              void* d_out, int out_size, void* d_ws,
                              size_t ws_size, hipStream_t stream) {
  const float* tensor_input = (const float*)d_in[0];
  const float* seed_ic   = (const float*)d_in[1];
  const float* seed_sir  = (const float*)d_in[2];
  const float* seed_icc  = (const float*)d_in[3];
  const float* seed_sirc = (const float*)d_in[4];
  const float* scalars   = (const float*)d_in[5];
  const float* seed_w1   = (const float*)d_in[6];
  const float* seed_b1   = (const float*)d_in[7];
  const float* seed_w2   = (const float*)d_in[8];
  const float* seed_b2   = (const float*)d_in[9];
  const float* scal_w1   = (const float*)d_in[10];
  const float* scal_b1   = (const float*)d_in[11];
  const float* scal_w2   = (const float*)d_in[12];
  const float* scal_b2   = (const float*)d_in[13];
  const float* cls_w1    = (const float*)d_in[14];
  const float* cls_b1    = (const float*)d_in[15];
  const float* cls_w2    = (const float*)d_in[16];
  const float* cls_b2    = (const float*)d_in[17];

  float* combined = (float*)d_ws;   // 1024 x 664 f32 = 2.72 MB
  float* outp = (float*)d_out;      // 1024 f32

  cp_als_kernel<<<1024, 256, 0, stream>>>(tensor_input, combined);
  seed_mlp_kernel<<<64, 128, 0, stream>>>(seed_ic, seed_sir, seed_icc,
                                          seed_sirc, seed_w1, seed_b1, seed_w2,
                                          seed_b2, combined);
  scal_mlp_kernel<<<64, 128, 0, stream>>>(scalars, scal_w1, scal_b1, scal_w2,
                                          scal_b2, combined);
  cls_kernel<<<64, 128, 0, stream>>>(combined, cls_w1, cls_b1, cls_w2, cls_b2,
                                     outp);
  (void)in_sizes; (void)n_in; (void)out_size; (void)ws_size;
}